// BoilerplateLoss_32014686224515
// MI455X (gfx1250) — compile-verified
//
#include <hip/hip_runtime.h>
#include <hip/hip_bf16.h>
#include <math.h>

// Problem constants (from the reference): C=1000 classes, K=10 attack idx/row.
#define NC 1000
#define NK 10
#define WAVES_PER_BLOCK 8

typedef __attribute__((ext_vector_type(4))) unsigned int u32x4;
typedef __attribute__((ext_vector_type(8))) int          i32x8;
typedef __attribute__((ext_vector_type(4))) int          i32x4;

__global__ __launch_bounds__(256) void attack_loss_kernel(
    const float* __restrict__ y_pred,
    const int*   __restrict__ y_attack,
    float*       __restrict__ out,
    int B)
{
    __shared__ float sm[WAVES_PER_BLOCK][NC];

    const int lane = threadIdx.x & 31;
    const int w    = __builtin_amdgcn_readfirstlane((int)(threadIdx.x >> 5));
    const int row  = __builtin_amdgcn_readfirstlane((int)(blockIdx.x * WAVES_PER_BLOCK + w));
    if (row >= B) return;

    const float* rowp = y_pred + (size_t)row * NC;
    const int*   ap   = y_attack + (size_t)row * NK;

    // gfx1250: prefetch the attack-index cacheline (global_prefetch_b8)
    __builtin_prefetch(ap, 0, 0);

    // ---- Stage this wave's row into its LDS slice via the Tensor Data Mover ----
#if __has_builtin(__builtin_amdgcn_tensor_load_to_lds)
    {
        unsigned long long ga = (unsigned long long)(uintptr_t)rowp;
        // Low 32 bits of a flat LDS pointer are the wave-relative LDS byte offset
        unsigned lds_addr = (unsigned)(unsigned long long)(uintptr_t)&sm[w][0];

        u32x4 g0;
        g0.x = 1u;                                   // count=1, gather off, user mode
        g0.y = lds_addr;                             // D#.lds_addr (bytes)
        g0.z = (unsigned)ga;                         // global_addr[31:0]
        g0.w = (unsigned)(ga >> 32) | (2u << 30);    // global_addr[56:32] | type=2

        i32x8 g1;
        g1[0] = (int)(2u << 16);                     // workgroup_mask=0, data_size=2 (4B)
        g1[1] = (int)((unsigned)NC << 16);           // tensor_dim0[15:0] in bits 63:48
        g1[2] = (int)(1u << 16);                     // tensor_dim0 hi=0 | tensor_dim1=1
        g1[3] = (int)((unsigned)NC << 16);           // tensor_dim1 hi=0 | tile_dim0=NC
        g1[4] = 1;                                   // tile_dim1=1 | tile_dim2=0
        g1[5] = NC;                                  // tensor_dim0_stride lo32
        g1[6] = 0;                                   // stride0 hi | stride1 lo
        g1[7] = 0;                                   // stride1 hi

        i32x4 gz4 = {0, 0, 0, 0};                    // groups 2/3 unused (<=2D tensor)
        i32x8 gz8 = {0, 0, 0, 0, 0, 0, 0, 0};        // extra group (clang-23 6-arg form)
        __builtin_amdgcn_tensor_load_to_lds(g0, g1, gz4, gz4, gz8, 0);
    }
#else
    for (int c = lane; c < NC; c += 32) sm[w][c] = rowp[c];
#endif

    // Attack indices: wave-uniform broadcast loads (overlap with the TDM transfer)
    int aidx[NK];
#pragma unroll
    for (int k = 0; k < NK; ++k) aidx[k] = ap[k];

#if __has_builtin(__builtin_amdgcn_s_wait_tensorcnt)
    __builtin_amdgcn_s_wait_tensorcnt(0);            // LDS slice now valid
#endif

    // ---- Pass 1: max over non-attack logits (branchless membership test) ----
    float mnot = -INFINITY;
    for (int c = lane; c < NC; c += 32) {
        float x = sm[w][c];
        bool att = false;
#pragma unroll
        for (int k = 0; k < NK; ++k) att |= (c == aidx[k]);
        mnot = fmaxf(mnot, att ? -INFINITY : x);
    }
#pragma unroll
    for (int off = 16; off; off >>= 1)
        mnot = fmaxf(mnot, __shfl_xor(mnot, off, 32));

    // Attack logits from LDS (same-address broadcast reads)
    float ax[NK];
    float mall = mnot;
#pragma unroll
    for (int k = 0; k < NK; ++k) {
        ax[k] = sm[w][aidx[k]];
        mall  = fmaxf(mall, ax[k]);
    }

    // ---- Pass 2: softmax denominator ----
    float s = 0.f;
    for (int c = lane; c < NC; c += 32) s += __expf(sm[w][c] - mall);
#pragma unroll
    for (int off = 16; off; off >>= 1) s += __shfl_xor(s, off, 32);

    // ---- Per-row tail (tiny, fp64 to match the reference's double upcast) ----
    if (lane == 0) {
        double invS = 1.0 / (double)s;

        double p[NK];
        double pmin = 1e300;
#pragma unroll
        for (int k = 0; k < NK; ++k) {
            p[k] = (double)__expf(ax[k] - mall) * invS;
            pmin = fmin(pmin, p[k]);
        }
        double pnot       = (double)__expf(mnot - mall) * invS;
        double macro_loss = pnot - pmin;

        // sorting: generalized mean (p=9) of 5 + 5*diff(in_att)
        double acc = 0.0;
#pragma unroll
        for (int j = 0; j < NK - 1; ++j) {
            double t  = 5.0 + 5.0 * (p[j + 1] - p[j]);
            double t2 = t * t;
            double t4 = t2 * t2;
            acc += t4 * t4 * t;                       // t^9
        }
        double s9      = pow(acc * (1.0 / (NK - 1)), 1.0 / 9.0);
        double sorting = (s9 - 5.0) * 0.2;

        // final: generalized mean (p=10) of the surjected pair
        double a0 = 5.0 + 5.0 * macro_loss;
        double a1 = 5.0 + 5.0 * sorting;
        double b0 = a0 * a0, b1 = a1 * a1;
        double a0_10 = b0 * b0 * b0 * b0 * b0;        // a0^10
        double a1_10 = b1 * b1 * b1 * b1 * b1;        // a1^10
        double g = pow(0.5 * (a0_10 + a1_10), 0.1);

        out[row] = (float)((g - 5.0) * 0.2);
    }
}

extern "C" void kernel_launch(void* const* d_in, const int* in_sizes, int n_in,
                              void* d_out, int out_size, void* d_ws, size_t ws_size,
                              hipStream_t stream) {
    const float* y_pred   = (const float*)d_in[0];
    const int*   y_attack = (const int*)d_in[1];
    float*       out      = (float*)d_out;

    const int B = in_sizes[0] / NC;                  // 32768
    const int blocks = (B + WAVES_PER_BLOCK - 1) / WAVES_PER_BLOCK;

    attack_loss_kernel<<<dim3(blocks), dim3(256), 0, stream>>>(y_pred, y_attack, out, B);
}